// InvariantMultiACE_4544075399637
// MI455X (gfx1250) — compile-verified
//
#include <hip/hip_runtime.h>
#include <hip/hip_bf16.h>
#include <math.h>

typedef __attribute__((ext_vector_type(2))) float v2f;
typedef __attribute__((ext_vector_type(4))) float v4f;
typedef __attribute__((ext_vector_type(8))) float v8f;

#define NUM_ELEM 10
#define HIDDEN   128
#define NUM_BESSEL 8
#define R_MAX 5.0f
#define GROUPS 64

// ---------------------------------------------------------------- e0 ------
__global__ void zero_e0_kernel(float* e0) {
    if (threadIdx.x < GROUPS) e0[threadIdx.x] = 0.0f;
}

__global__ void e0_kernel(const float* __restrict__ node_attrs,
                          const float* __restrict__ atomic_energies,
                          const int*   __restrict__ batch,
                          float* __restrict__ e0, int n_nodes) {
    __shared__ float bins[GROUPS];
    const int t = threadIdx.x;
    if (t < GROUPS) bins[t] = 0.0f;
    __syncthreads();
    const int i = blockIdx.x * blockDim.x + t;
    if (i < n_nodes) {
        float s = 0.0f;
#pragma unroll
        for (int k = 0; k < NUM_ELEM; ++k)
            s += node_attrs[i * NUM_ELEM + k] * atomic_energies[k];
        atomicAdd(&bins[batch[i]], s);
    }
    __syncthreads();
    if (t < GROUPS) atomicAdd(&e0[t], bins[t]);
}

// ------------------------------------------------ node_feats GEMM (WMMA) --
// C[N x 128] = A[N x 10] * B[10 x 128], f32, K padded 10 -> 12 (3 x K4 WMMA).
// One block = 8 wave32; wave w owns 16x16 tile (rows blockIdx.x*16, cols w*16).
// Out-of-range K handled with clamped (in-bounds) loads * {0,1} mask so there
// is no exec-mask divergence around the WMMA sequence.
__global__ void node_feats_wmma_kernel(const float* __restrict__ node_attrs,
                                       const float* __restrict__ emb_W,
                                       float* __restrict__ node_feats) {
    const int lane = threadIdx.x & 31;
    const int wave = threadIdx.x >> 5;        // 0..7 -> column tile
    const int m0   = blockIdx.x * 16;         // node row base (N % 16 == 0)
    const int n0   = wave * 16;               // hidden col base
    const int half = lane >> 4;               // 0: K pair {0,1}; 1: K pair {2,3}
    const int l16  = lane & 15;

    const int arow = m0 + l16;                // A row for this lane
    const int bcol = n0 + l16;                // B col for this lane

    v8f acc = {};
#pragma unroll
    for (int kk = 0; kk < 12; kk += 4) {
        const int k0  = kk + half * 2;
        const int k0c = (k0 < NUM_ELEM - 1) ? k0 : (NUM_ELEM - 2); // k0c,k0c+1 in-bounds
        const float w0 = (k0     < NUM_ELEM) ? 1.0f : 0.0f;
        const float w1 = (k0 + 1 < NUM_ELEM) ? 1.0f : 0.0f;
        // A operand: lane l16 = row M, half selects K pair (ISA 16x4 f32 layout)
        v2f a;
        a.x = node_attrs[arow * NUM_ELEM + k0c    ] * w0;
        a.y = node_attrs[arow * NUM_ELEM + k0c + 1] * w1;
        // B operand: lane l16 = col N, half selects K pair (4x16 f32 layout)
        v2f b;
        b.x = emb_W[(k0c    ) * HIDDEN + bcol] * w0;
        b.y = emb_W[(k0c + 1) * HIDDEN + bcol] * w1;
        // 8 args: (neg_a, A, neg_b, B, c_mod, C, reuse_a, reuse_b)
        acc = __builtin_amdgcn_wmma_f32_16x16x4_f32(false, a, false, b,
                                                    (short)0, acc, false, false);
    }
    // D layout: VGPR r, lanes 0-15 -> row r, lanes 16-31 -> row r+8; col = l16
#pragma unroll
    for (int r = 0; r < 8; ++r) {
        const int row = m0 + r + half * 8;
        __builtin_nontemporal_store(acc[r], node_feats + row * HIDDEN + bcol);
    }
}

// ---------------------------------------------------------------- edges ---
__global__ void edge_kernel(const float* __restrict__ positions,
                            const float* __restrict__ shifts,
                            const int*   __restrict__ edge_index,
                            float* __restrict__ edge_attrs,
                            float* __restrict__ edge_feats, int n_edges) {
    const int e = blockIdx.x * blockDim.x + threadIdx.x;
    if (e >= n_edges) return;

    const int s = edge_index[e];
    const int r = edge_index[n_edges + e];

    const float vx = positions[3 * r + 0] - positions[3 * s + 0] + shifts[3 * e + 0];
    const float vy = positions[3 * r + 1] - positions[3 * s + 1] + shifts[3 * e + 1];
    const float vz = positions[3 * r + 2] - positions[3 * s + 2] + shifts[3 * e + 2];

    const float len2 = vx * vx + vy * vy + vz * vz;
    const float len  = sqrtf(len2);
    const float inv  = 1.0f / len;
    const float x = vx * inv, y = vy * inv, z = vz * inv;
    const float x2 = x * x, y2 = y * y, z2 = z * z;

    // --- spherical harmonics (16 components) ---
    const float c3  = 1.7320508075688772f;   // sqrt(3)
    const float c5  = 2.23606797749979f;     // sqrt(5)
    const float c7  = 2.6457513110645907f;   // sqrt(7)
    const float c15 = 3.872983346207417f;    // sqrt(15)
    const float c42 = 6.48074069840786f;     // sqrt(42)
    const float c70 = 8.366600265340756f;    // sqrt(70)
    const float c105 = 10.246950765959598f;  // sqrt(105)

    v4f sh0, sh1, sh2, sh3;
    sh0.x = 1.0f;
    sh0.y = c3 * x;
    sh0.z = c3 * y;
    sh0.w = c3 * z;
    sh1.x = c15 * x * y;
    sh1.y = c15 * y * z;
    sh1.z = 0.5f * c5 * (3.0f * z2 - 1.0f);
    sh1.w = c15 * x * z;
    sh2.x = 0.5f * c15 * (x2 - y2);
    sh2.y = 0.25f * c70 * y * (3.0f * x2 - y2);
    sh2.z = c105 * x * y * z;
    sh2.w = 0.25f * c42 * y * (5.0f * z2 - 1.0f);
    sh3.x = 0.5f * c7 * z * (5.0f * z2 - 3.0f);
    sh3.y = 0.25f * c42 * x * (5.0f * z2 - 1.0f);
    sh3.z = 0.5f * c105 * z * (x2 - y2);
    sh3.w = 0.25f * c70 * x * (x2 - 3.0f * y2);

    v4f* ea = (v4f*)(edge_attrs + (size_t)e * 16);
    __builtin_nontemporal_store(sh0, ea + 0);
    __builtin_nontemporal_store(sh1, ea + 1);
    __builtin_nontemporal_store(sh2, ea + 2);
    __builtin_nontemporal_store(sh3, ea + 3);

    // --- radial embedding: bessel * polynomial cutoff envelope ---
    const float xr = len / R_MAX;
    const float xr2 = xr * xr;
    const float xr4 = xr2 * xr2;
    const float xr5 = xr4 * xr;
    const float xr6 = xr5 * xr;
    const float xr7 = xr6 * xr;
    // p=5: 1 - 21 xr^5 + 35 xr^6 - 15 xr^7
    float env = 1.0f - 21.0f * xr5 + 35.0f * xr6 - 15.0f * xr7;
    env = (xr < 1.0f) ? env : 0.0f;

    const float pref = 0.6324555320336759f * inv * env; // sqrt(2/R_MAX)/r * env
    const float w = 3.14159265358979323846f * len / R_MAX;

    v4f ef0, ef1;
    ef0.x = pref * __sinf(1.0f * w);
    ef0.y = pref * __sinf(2.0f * w);
    ef0.z = pref * __sinf(3.0f * w);
    ef0.w = pref * __sinf(4.0f * w);
    ef1.x = pref * __sinf(5.0f * w);
    ef1.y = pref * __sinf(6.0f * w);
    ef1.z = pref * __sinf(7.0f * w);
    ef1.w = pref * __sinf(8.0f * w);

    v4f* ef = (v4f*)(edge_feats + (size_t)e * 8);
    __builtin_nontemporal_store(ef0, ef + 0);
    __builtin_nontemporal_store(ef1, ef + 1);
}

// --------------------------------------------------------------- launch ---
extern "C" void kernel_launch(void* const* d_in, const int* in_sizes, int n_in,
                              void* d_out, int out_size, void* d_ws, size_t ws_size,
                              hipStream_t stream) {
    const float* positions       = (const float*)d_in[0];
    const float* shifts          = (const float*)d_in[1];
    const float* node_attrs      = (const float*)d_in[2];
    const float* emb_W           = (const float*)d_in[3];
    const float* atomic_energies = (const float*)d_in[4];
    const int*   edge_index      = (const int*)d_in[5];
    const int*   batch           = (const int*)d_in[6];

    const int N = in_sizes[0] / 3;       // positions: N x 3
    const int E = in_sizes[5] / 2;       // edge_index: 2 x E

    float* out        = (float*)d_out;
    float* e0         = out;                                   // [64]
    float* node_feats = out + GROUPS;                          // [N x 128]
    float* edge_attrs = node_feats + (size_t)N * HIDDEN;       // [E x 16]
    float* edge_feats = edge_attrs + (size_t)E * 16;           // [E x 8]

    zero_e0_kernel<<<1, GROUPS, 0, stream>>>(e0);
    e0_kernel<<<(N + 255) / 256, 256, 0, stream>>>(node_attrs, atomic_energies,
                                                   batch, e0, N);
    node_feats_wmma_kernel<<<(N + 15) / 16, 256, 0, stream>>>(node_attrs, emb_W,
                                                              node_feats);
    edge_kernel<<<(E + 255) / 256, 256, 0, stream>>>(positions, shifts, edge_index,
                                                     edge_attrs, edge_feats, E);
}